// BERTForContrastiveLearningForTokenMetric_89464168776514
// MI455X (gfx1250) — compile-verified
//
#include <hip/hip_runtime.h>
#include <math.h>

typedef __attribute__((ext_vector_type(2))) float v2f;
typedef __attribute__((ext_vector_type(8))) float v8f;

#define B_DIM 64
#define S_DIM 512
#define D_DIM 768
#define EPS 1e-8f

// ws layout (floats):
// [0..767]  proto accumulator -> finalized proto
// [768]     entity-mask count
// [769]     nonzero-label count
// [770]     ||proto||
// [771]     sum of sims*nz
#define WS_ENTCNT 768
#define WS_NZCNT  769
#define WS_PNORM  770
#define WS_LOSS   771
#define WS_FLOATS 772

// ---------------- kernel 0: zero the workspace accumulators ----------------
__global__ void k0_zero_ws(float* ws) {
    int i = blockIdx.x * blockDim.x + threadIdx.x;
    if (i < WS_FLOATS) ws[i] = 0.0f;
}

// ---------------- kernel 1: proto accumulation + counts --------------------
// One block per batch row b. 256 threads each own 3 columns of D=768.
// Labels are staged into LDS; only masked rows (label==ent && s!=0) are read
// from global, so this pass touches only ~10% of the 96MB tensor.
__global__ void k1_proto_accum(const float* __restrict__ logits,
                               const int* __restrict__ labels,
                               const int* __restrict__ entity_id,
                               float* __restrict__ ws) {
    __shared__ int slab[S_DIM];
    const int b   = blockIdx.x;
    const int tid = threadIdx.x;
    const int ent = entity_id[0];

    const int* lb = labels + b * S_DIM;
    slab[tid]       = lb[tid];
    slab[tid + 256] = lb[tid + 256];
    __syncthreads();

    const float* base = logits + (size_t)b * S_DIM * D_DIM;
    float a0 = 0.f, a1 = 0.f, a2 = 0.f;
    int entc = 0, nzc = 0;

    for (int s = 0; s < S_DIM; ++s) {
        int lab = slab[s];
        if (tid == 0) {
            if (lab != 0) nzc++;
            if (lab == ent && s != 0) entc++;
        }
        if (lab == ent && s != 0) {
            const float* r = base + (size_t)s * D_DIM;
            a0 += r[tid];
            a1 += r[tid + 256];
            a2 += r[tid + 512];
        }
    }
    atomicAdd(ws + tid,       a0);
    atomicAdd(ws + tid + 256, a1);
    atomicAdd(ws + tid + 512, a2);
    if (tid == 0) {
        atomicAdd(ws + WS_ENTCNT, (float)entc);
        atomicAdd(ws + WS_NZCNT,  (float)nzc);
    }
}

// ---------------- kernel 2: finalize proto, compute its norm ---------------
__global__ void k2_proto_finalize(float* __restrict__ ws) {
    __shared__ float red[256];
    const int tid = threadIdx.x;
    const float n = fmaxf(ws[WS_ENTCNT], 1.0f);
    float p0 = ws[tid]       / n;
    float p1 = ws[tid + 256] / n;
    float p2 = ws[tid + 512] / n;
    ws[tid]       = p0;
    ws[tid + 256] = p1;
    ws[tid + 512] = p2;
    red[tid] = p0 * p0 + p1 * p1 + p2 * p2;
    __syncthreads();
    for (int off = 128; off > 0; off >>= 1) {
        if (tid < off) red[tid] += red[tid + off];
        __syncthreads();
    }
    if (tid == 0) ws[WS_PNORM] = sqrtf(red[0]);
}

// ---------------- kernel 3: streaming WMMA matvec + norms + sims -----------
// 256 blocks x 256 threads (8 waves). Each wave owns 16 token rows and runs
// 192 V_WMMA_F32_16X16X4_F32 ops (K=768 in chunks of 4). proto is replicated
// across all 16 B-columns so every column of the D tile is the per-token dot.
// A-matrix lane layout (32-bit 16x4): lane l<16 -> (M=l, K=0,1); lane l>=16
// -> (M=l-16, K=2,3). The same float2 loads feed the ||x||^2 accumulation.
__global__ void k3_sims_wmma(const float* __restrict__ logits,
                             const int* __restrict__ labels,
                             float* __restrict__ ws) {
    __shared__ __align__(16) float sproto[D_DIM];
    const int tid = threadIdx.x;
    // stage proto into LDS (3 elements per thread)
    sproto[tid]       = ws[tid];
    sproto[tid + 256] = ws[tid + 256];
    sproto[tid + 512] = ws[tid + 512];
    __syncthreads();

    const float pnorm = ws[WS_PNORM];

    const int wave = tid >> 5;
    const int lane = tid & 31;
    const int half = lane >> 4;
    const int m    = lane & 15;

    const int baseTok = blockIdx.x * 128 + wave * 16;
    const int rowTok  = baseTok + m;                 // this lane's token row
    const float* row  = logits + (size_t)rowTok * D_DIM;
    const int koff    = 2 * half;                    // K phase within chunk

    v8f acc = {};
    float sumsq = 0.f;

#pragma unroll 4
    for (int k0 = 0; k0 < D_DIM; k0 += 4) {
        v2f a = *(const v2f*)(row + k0 + koff);          // A: 2 VGPRs, global
        v2f b = *(const v2f*)(&sproto[k0 + koff]);       // B: 2 VGPRs, LDS
        // D = A x B + C   (fp32 WMMA, no precision loss)
        acc = __builtin_amdgcn_wmma_f32_16x16x4_f32(
            /*neg_a=*/false, a, /*neg_b=*/false, b,
            /*c_mod=*/(short)0, acc, /*reuse_a=*/false, /*reuse_b=*/false);
        sumsq = fmaf(a.x, a.x, sumsq);
        sumsq = fmaf(a.y, a.y, sumsq);
    }

    // combine half-wave partials: lane l holds K%4 in {0,1}, lane l^16 holds {2,3}
    sumsq += __shfl_xor(sumsq, 16, 32);

    const int lab = labels[rowTok];
    const float norm  = sqrtf(sumsq);
    const float denom = fmaxf(norm * pnorm, EPS);

    // Extract per-token dots from D tile: acc[r] on lanes 0-15 is M=r,
    // on lanes 16-31 is M=r+8 (all 16 columns identical). Route token t's
    // dot to lane t (t<16) via lane-0 / lane-16 broadcasts.
    float dot = 0.f;
#pragma unroll
    for (int r = 0; r < 8; ++r) {
        float vlo = __shfl(acc[r], 0,  32);  // dot of token baseTok + r
        float vhi = __shfl(acc[r], 16, 32);  // dot of token baseTok + 8 + r
        if (lane == r)     dot = vlo;
        if (lane == r + 8) dot = vhi;
    }

    float c = (lane < 16 && lab != 0) ? (dot / denom) : 0.f;

    // wave-level sum, then one atomic per wave
    for (int off = 16; off > 0; off >>= 1) c += __shfl_down(c, off, 32);
    if (lane == 0) atomicAdd(ws + WS_LOSS, c);
}

// ---------------- kernel 4: scalar finalize --------------------------------
__global__ void k4_finalize(const float* __restrict__ ws, float* __restrict__ out) {
    out[0] = ws[WS_LOSS] / fmaxf(ws[WS_NZCNT], 1.0f);
}

extern "C" void kernel_launch(void* const* d_in, const int* in_sizes, int n_in,
                              void* d_out, int out_size, void* d_ws, size_t ws_size,
                              hipStream_t stream) {
    const float* logits = (const float*)d_in[0];
    const int*   labels = (const int*)d_in[1];
    const int*   entity = (const int*)d_in[2];
    float* ws  = (float*)d_ws;
    float* out = (float*)d_out;

    k0_zero_ws<<<4, 256, 0, stream>>>(ws);
    k1_proto_accum<<<B_DIM, 256, 0, stream>>>(logits, labels, entity, ws);
    k2_proto_finalize<<<1, 256, 0, stream>>>(ws);
    // 64*512 = 32768 tokens, 16 tokens/wave, 8 waves/block -> 256 blocks
    k3_sims_wmma<<<256, 256, 0, stream>>>(logits, labels, ws);
    k4_finalize<<<1, 1, 0, stream>>>(ws, out);
}